// QuantumRegression__gen019_65481071398164
// MI455X (gfx1250) — compile-verified
//
#include <hip/hip_runtime.h>

typedef __attribute__((ext_vector_type(2))) float v2f;
typedef __attribute__((ext_vector_type(8))) float v8f;

#define NW 10
#define WAVES_PER_BLOCK 4
#define SAMPLES_PER_WAVE 4

// CNOT-ladder permutation on a 5-bit group: wire j <-> bit (4-j).
// forward: CNOT(j,j+1) for j=0..3 applied in order.
__device__ __forceinline__ int sigma5(int v) {
  #pragma unroll
  for (int p = 4; p >= 1; --p)
    if ((v >> p) & 1) v ^= (1 << (p - 1));
  return v;
}
__device__ __forceinline__ int sigma5inv(int v) {
  #pragma unroll
  for (int p = 1; p <= 4; ++p)
    if ((v >> p) & 1) v ^= (1 << (p - 1));
  return v;
}
// stored-row order: even logical rows first. invQ(r) = stored slot of logical row r
__device__ __forceinline__ int invQ(int r) { return (r >> 1) + ((r & 1) << 4); }

struct SharedData {
  float Hre[2][1024];  // layer hi operator, rows/cols in stored (even-first) order
  float Him[2][1024];
  float Mre[2][1024];  // M_A^T  (B operand for the lo matmul, even rows)
  float Mim[2][1024];
  float u_re[2][10][2][2];
  float u_im[2][10][2][2];
  int   permB[32];     // column map: M_B^T[:,j] = M_A^T[:,permB[j]]
  float Sre[WAVES_PER_BLOCK][1024];
  float Sim[WAVES_PER_BLOCK][1024];
  float Tre[WAVES_PER_BLOCK][1024];
  float Tim[WAVES_PER_BLOCK][1024];
};

__device__ __forceinline__ v8f wmma4(v2f a, v2f b, v8f c) {
  // D = A(16x4) * B(4x16) + C, full f32
  return __builtin_amdgcn_wmma_f32_16x16x4_f32(false, a, false, b, (short)0, c,
                                               false, false);
}

// 32x32 complex GEMM: D = A * B.  A = state-or-H (rows striped as WMMA A operand),
// B = state-or-M^T.  BIMAG: whether B has an imaginary part.
// COLPERM: for the lo matmul, output tile rows ti=1 (odd logical rows) fetch
// permuted columns of B (== using M_B^T instead of M_A^T).
template <bool BIMAG, bool COLPERM>
__device__ __forceinline__ void cgemm32(const float* Ar, const float* Ai,
                                        const float* Br, const float* Bi,
                                        float* Dr, float* Di, const int* permB,
                                        int lane) {
  const int half = lane >> 4;
  const int l15  = lane & 15;
  #pragma unroll
  for (int ti = 0; ti < 2; ++ti) {
    // Preload this lane's A-row chunks (ISA A layout: lane=M, VGPR0/1 = K pair,
    // lane halves cover K pairs {4k,4k+1} vs {4k+2,4k+3})
    v2f ar[8], ai[8], nai[8];
    const int m = ti * 16 + l15;
    #pragma unroll
    for (int kk = 0; kk < 8; ++kk) {
      const int kb = kk * 4 + 2 * half;
      ar[kk].x  = Ar[m * 32 + kb];
      ar[kk].y  = Ar[m * 32 + kb + 1];
      ai[kk].x  = Ai[m * 32 + kb];
      ai[kk].y  = Ai[m * 32 + kb + 1];
      nai[kk]   = -ai[kk];
    }
    #pragma unroll
    for (int tj = 0; tj < 2; ++tj) {
      const int n    = tj * 16 + l15;
      const int ncol = (COLPERM && ti) ? permB[n] : n;
      v8f accR = {0.f, 0.f, 0.f, 0.f, 0.f, 0.f, 0.f, 0.f};
      v8f accI = {0.f, 0.f, 0.f, 0.f, 0.f, 0.f, 0.f, 0.f};
      #pragma unroll
      for (int kk = 0; kk < 8; ++kk) {
        const int kb = kk * 4 + 2 * half;
        v2f br;
        br.x = Br[kb * 32 + ncol];
        br.y = Br[(kb + 1) * 32 + ncol];
        accR = wmma4(ar[kk], br, accR);   // Re += Ar*Br
        accI = wmma4(ai[kk], br, accI);   // Im += Ai*Br
        if (BIMAG) {
          v2f bi;
          bi.x = Bi[kb * 32 + ncol];
          bi.y = Bi[(kb + 1) * 32 + ncol];
          accR = wmma4(nai[kk], bi, accR);  // Re -= Ai*Bi
          accI = wmma4(ar[kk], bi, accI);   // Im += Ar*Bi
        }
      }
      // D layout: VGPR v -> row v (lanes 0-15) / v+8 (lanes 16-31)
      #pragma unroll
      for (int v = 0; v < 8; ++v) {
        const int row = ti * 16 + v + 8 * half;
        const int col = tj * 16 + l15;
        Dr[row * 32 + col] = accR[v];
        Di[row * 32 + col] = accI[v];
      }
    }
  }
}

__global__ __launch_bounds__(128) void qnn_quantum_wmma_kernel(
    const float* __restrict__ x, const float* __restrict__ params,
    const float* __restrict__ head_w, const float* __restrict__ head_b,
    float* __restrict__ out, int B) {
  __shared__ SharedData sh;
  const int tid  = threadIdx.x;
  const int lane = tid & 31;
  const int wave = tid >> 5;

  // ---------------- per-block precompute (batch-invariant) ----------------
  if (tid < 20) {
    const int l = tid / 10, w = tid % 10;
    const float phi = params[(l * NW + w) * 3 + 0];
    const float th  = params[(l * NW + w) * 3 + 1];
    const float om  = params[(l * NW + w) * 3 + 2];
    const float c = cosf(0.5f * th), s = sinf(0.5f * th);
    const float ap = 0.5f * (phi + om), am = 0.5f * (phi - om);
    const float cp = cosf(ap), sp = sinf(ap);
    const float cm = cosf(am), sm = sinf(am);
    sh.u_re[l][w][0][0] =  c * cp;  sh.u_im[l][w][0][0] = -c * sp;  // c e^{-i(φ+ω)/2}
    sh.u_re[l][w][0][1] = -s * cm;  sh.u_im[l][w][0][1] = -s * sm;  // -s e^{+i(φ-ω)/2}
    sh.u_re[l][w][1][0] =  s * cm;  sh.u_im[l][w][1][0] = -s * sm;  //  s e^{-i(φ-ω)/2}
    sh.u_re[l][w][1][1] =  c * cp;  sh.u_im[l][w][1][1] =  c * sp;  //  c e^{+i(φ+ω)/2}
  }
  if (tid < 32) sh.permB[tid] = sigma5(sigma5inv(tid) ^ 16);
  __syncthreads();

  // Build H_st (hi Kronecker, CNOT row-perm + even-first reordering folded in)
  // and M_A^T (lo Kronecker with CNOT col-perm folded in), both layers.
  for (int id = tid; id < 4096; id += 128) {
    const int mat = id >> 10;        // 0..3
    const int l   = mat >> 1;
    const int kind = mat & 1;        // 0 = H, 1 = M^T
    const int i = (id >> 5) & 31;    // source row index
    const int j = id & 31;           // source col index
    const int wb = kind ? 5 : 0;
    float vr = 1.f, vi = 0.f;
    #pragma unroll
    for (int q = 0; q < 5; ++q) {
      const int bi = (i >> (4 - q)) & 1;
      const int bj = (j >> (4 - q)) & 1;
      const float ur = sh.u_re[l][wb + q][bi][bj];
      const float ui = sh.u_im[l][wb + q][bi][bj];
      const float nr = vr * ur - vi * ui;
      vi = vr * ui + vi * ur;
      vr = nr;
    }
    if (kind == 0) {
      // H = P_hi * (⊗Rot); stored with even-first perm on rows AND cols
      const int dr = invQ(sigma5(i));
      const int dc = invQ(j);
      sh.Hre[l][dr * 32 + dc] = vr;
      sh.Him[l][dr * 32 + dc] = vi;
    } else {
      // U_lo[i=c][j=k] -> M_A^T[k][sigma_lo(c)]   (M_A = P_lo * U_lo)
      const int dc = sigma5(i);
      sh.Mre[l][j * 32 + dc] = vr;
      sh.Mim[l][j * 32 + dc] = vi;
    }
  }
  __syncthreads();

  // head weights in registers (unrolled constant indexing)
  float hw[20];
  #pragma unroll
  for (int k = 0; k < 20; ++k) hw[k] = head_w[k];
  const float hb = head_b[0];

  float* Sre = sh.Sre[wave];
  float* Sim = sh.Sim[wave];
  float* Tre = sh.Tre[wave];
  float* Tim = sh.Tim[wave];

  const int base = (blockIdx.x * WAVES_PER_BLOCK + wave) * SAMPLES_PER_WAVE;
  for (int it = 0; it < SAMPLES_PER_WAVE; ++it) {
    const int b = base + it;
    if (b >= B) break;
    if (b + 1 < B) __builtin_prefetch(&x[(b + 1) * NW], 0, 1);

    // -------- encoding: S = hi(r) * lo(c), rows stored even-first ----------
    float cw[10], sw[10];
    #pragma unroll
    for (int w = 0; w < 10; ++w) {
      const float xv = x[b * NW + w];
      cw[w] = cosf(0.5f * xv);
      sw[w] = sinf(0.5f * xv);
    }
    float lov = 1.f;
    #pragma unroll
    for (int q = 0; q < 5; ++q)
      lov *= (((lane >> (4 - q)) & 1) ? sw[5 + q] : cw[5 + q]);
    const int r = (lane < 16) ? 2 * lane : 2 * (lane - 16) + 1;  // logical row
    float hiv = 1.f;
    #pragma unroll
    for (int q = 0; q < 5; ++q)
      hiv *= (((r >> (4 - q)) & 1) ? sw[q] : cw[q]);
    #pragma unroll
    for (int c = 0; c < 32; ++c) {
      const float lc = __shfl(lov, c, 32);
      Sre[lane * 32 + c] = hiv * lc;   // imag part identically zero -> skipped
    }
    asm volatile("s_wait_dscnt 0" ::: "memory");

    // -------- layer 1: hi (B imag zero), then lo (full complex) -----------
    cgemm32<false, false>(sh.Hre[0], sh.Him[0], Sre, Sim, Tre, Tim, sh.permB, lane);
    asm volatile("s_wait_dscnt 0" ::: "memory");
    cgemm32<true, true>(Tre, Tim, sh.Mre[0], sh.Mim[0], Sre, Sim, sh.permB, lane);
    asm volatile("s_wait_dscnt 0" ::: "memory");
    // -------- layer 2 ------------------------------------------------------
    cgemm32<true, false>(sh.Hre[1], sh.Him[1], Sre, Sim, Tre, Tim, sh.permB, lane);
    asm volatile("s_wait_dscnt 0" ::: "memory");
    cgemm32<true, true>(Tre, Tim, sh.Mre[1], sh.Mim[1], Sre, Sim, sh.permB, lane);
    asm volatile("s_wait_dscnt 0" ::: "memory");

    // -------- features + head, fused ---------------------------------------
    float rre[32], rim[32];
    #pragma unroll
    for (int c = 0; c < 32; ++c) {
      rre[c] = Sre[lane * 32 + c];
      rim[c] = Sim[lane * 32 + c];
    }
    float acc = 0.f;
    float prow = 0.f;
    float zlo[5] = {0.f, 0.f, 0.f, 0.f, 0.f};
    #pragma unroll
    for (int c = 0; c < 32; ++c) {
      const float p = rre[c] * rre[c] + rim[c] * rim[c];
      prow += p;
      #pragma unroll
      for (int q = 0; q < 5; ++q)
        zlo[q] += (((c >> (4 - q)) & 1) ? -p : p);
    }
    #pragma unroll
    for (int w = 0; w < 5; ++w)
      acc += hw[w] * ((((r >> (4 - w)) & 1) ? -prow : prow));
    #pragma unroll
    for (int q = 0; q < 5; ++q) acc += hw[5 + q] * zlo[q];
    // <X> on lo wires: within-row pairs
    #pragma unroll
    for (int q = 0; q < 5; ++q) {
      const int m = 1 << (4 - q);
      float sx = 0.f;
      #pragma unroll
      for (int c = 0; c < 32; ++c)
        if (!(c & m)) sx += rre[c] * rre[c | m] + rim[c] * rim[c | m];
      acc += hw[15 + q] * (2.f * sx);
    }
    // <X> on hi wires: partner row from LDS (stored even-first)
    #pragma unroll
    for (int w = 0; w < 5; ++w) {
      const int m = 1 << (4 - w);
      float sx = 0.f;
      if (!(r & m)) {
        const int s2 = invQ(r | m);
        #pragma unroll
        for (int c = 0; c < 32; ++c)
          sx += rre[c] * Sre[s2 * 32 + c] + rim[c] * Sim[s2 * 32 + c];
      }
      acc += hw[10 + w] * (2.f * sx);
    }
    // wave32 reduction over rows
    #pragma unroll
    for (int off = 16; off > 0; off >>= 1) acc += __shfl_down(acc, off, 32);
    if (lane == 0) out[b] = acc + hb;
  }
}

extern "C" void kernel_launch(void* const* d_in, const int* in_sizes, int n_in,
                              void* d_out, int out_size, void* d_ws, size_t ws_size,
                              hipStream_t stream) {
  const float* x      = (const float*)d_in[0];
  const float* params = (const float*)d_in[1];
  const float* head_w = (const float*)d_in[2];
  const float* head_b = (const float*)d_in[3];
  float* out          = (float*)d_out;
  const int B = in_sizes[0] / NW;
  const int samples_per_block = WAVES_PER_BLOCK * SAMPLES_PER_WAVE;
  const int blocks = (B + samples_per_block - 1) / samples_per_block;
  qnn_quantum_wmma_kernel<<<blocks, WAVES_PER_BLOCK * 32, 0, stream>>>(
      x, params, head_w, head_b, out, B);
}